// FNPGNNv8_E2_38019050504252
// MI455X (gfx1250) — compile-verified
//
#include <hip/hip_runtime.h>
#include <hip/hip_bf16.h>

// ---------------------------------------------------------------------------
// FNPGNN forward on gfx1250 (MI455X).  All dense GEMMs use
// V_WMMA_F32_16X16X4_F32 (fp32 in / fp32 accumulate on the matrix pipe),
// attention uses a fused online-softmax with Wh[b] staged in LDS.
// ---------------------------------------------------------------------------

typedef float v2f __attribute__((ext_vector_type(2)));
typedef float v8f __attribute__((ext_vector_type(8)));

#define NB   512
#define NN   116
#define NIN  125
#define HID  128
#define NH   4
#define DD   32
#define BN   (NB * NN)          // 59392 rows, divisible by 16
#define FLATD 1280
#define C1D  256

__device__ __forceinline__ v8f wmma4(v2f a, v2f b, v8f c) {
  // D = A(16x4 f32) x B(4x16 f32) + C(16x16 f32)
  return __builtin_amdgcn_wmma_f32_16x16x4_f32(
      /*neg_a=*/false, a, /*neg_b=*/false, b,
      /*c_mod=*/(short)0, c, /*reuse_a=*/false, /*reuse_b=*/false);
}

__device__ __forceinline__ float elu_f(float v) {
  return v > 0.f ? v : __expf(v) - 1.f;
}

// ---------------------------------------------------------------------------
// Encoder: h = bn(elu(x @ W_enc + b_enc)) ; x: (BN,125) W: (125,128)
// one block = 16 rows x 128 cols, 8 waves, wave w -> col tile w
// ---------------------------------------------------------------------------
__global__ __launch_bounds__(256) void k_encoder(
    const float* __restrict__ x, const float* __restrict__ W,
    const float* __restrict__ bias, const float* __restrict__ g,
    const float* __restrict__ beta, float* __restrict__ h) {
  __shared__ float As[16 * 128];
  const int row0 = blockIdx.x * 16;
  const int tid = threadIdx.x;
  for (int e = tid; e < 16 * 128; e += 256) {
    int m = e >> 7, k = e & 127;
    As[e] = (k < NIN) ? x[(size_t)(row0 + m) * NIN + k] : 0.f;
  }
  __syncthreads();
  const int wave = tid >> 5, lane = tid & 31;
  const int n0 = wave * 16;
  const int nl = lane & 15, ml = lane & 15, half = lane >> 4;
  v8f acc = {};
  for (int k0 = 0; k0 < 128; k0 += 4) {
    int ka = k0 + 2 * half;
    v2f a; a.x = As[ml * 128 + ka]; a.y = As[ml * 128 + ka + 1];
    v2f bf;
    bf.x = (ka     < NIN) ? W[ka * HID + n0 + nl]       : 0.f;
    bf.y = (ka + 1 < NIN) ? W[(ka + 1) * HID + n0 + nl] : 0.f;
    acc = wmma4(a, bf, acc);
  }
  const int n = n0 + nl;
  const float gs = g[n] * rsqrtf(1.f + 1e-5f);
  const float bb = beta[n], be = bias[n];
  const int mbase = half * 8;
#pragma unroll
  for (int r = 0; r < 8; ++r) {
    float v = acc[r] + be;
    v = elu_f(v);
    v = v * gs + bb;
    h[(size_t)(row0 + mbase + r) * HID + n] = v;
  }
}

// ---------------------------------------------------------------------------
// Wh = A(BN,128) @ B(128,128)  (gat weight), raw output
// ---------------------------------------------------------------------------
__global__ __launch_bounds__(256) void k_gemm128(
    const float* __restrict__ A, const float* __restrict__ Bm,
    float* __restrict__ C) {
  __shared__ float As[16 * 128];
  const int row0 = blockIdx.x * 16;
  const int tid = threadIdx.x;
  for (int e = tid; e < 16 * 128; e += 256)
    As[e] = A[(size_t)row0 * HID + e];
  __syncthreads();
  const int wave = tid >> 5, lane = tid & 31;
  const int n0 = wave * 16;
  const int nl = lane & 15, ml = lane & 15, half = lane >> 4;
  v8f acc = {};
  for (int k0 = 0; k0 < 128; k0 += 4) {
    int ka = k0 + 2 * half;
    v2f a; a.x = As[ml * 128 + ka]; a.y = As[ml * 128 + ka + 1];
    v2f bf; bf.x = Bm[ka * HID + n0 + nl]; bf.y = Bm[(ka + 1) * HID + n0 + nl];
    acc = wmma4(a, bf, acc);
  }
  const int n = n0 + nl;
  const int mbase = half * 8;
#pragma unroll
  for (int r = 0; r < 8; ++r)
    C[(size_t)(row0 + mbase + r) * HID + n] = acc[r];
}

// ---------------------------------------------------------------------------
// GAT attention, fused: one block per batch item.  Wh[b] in LDS, per-(n,h)
// src/dst projections in LDS, each wave does online-softmax over j for one
// (i,h), lanes carry the 32-wide head dim.  Epilogue: bn + elu + Wh_flat
// (+ residual) written in place into h.
// ---------------------------------------------------------------------------
__global__ __launch_bounds__(256) void k_attn(
    const float* __restrict__ Wh, const float* __restrict__ adj,
    const float* __restrict__ asrc, const float* __restrict__ adst,
    const float* __restrict__ gamma, const float* __restrict__ beta,
    float* __restrict__ h, int residual) {
  __shared__ float WhS[NN * HID];    // 59392 B
  __shared__ float srcS[NN * NH];    // 1856 B
  __shared__ float dstS[NN * NH];    // 1856 B
  const int b = blockIdx.x;
  const int tid = threadIdx.x;
  const float* Whb = Wh + (size_t)b * NN * HID;
  for (int e = tid; e < NN * HID; e += 256) WhS[e] = Whb[e];
  __syncthreads();
  // src[n,h] = Wh[n,h,:] . asrc ; dst likewise
  for (int t = tid; t < NN * NH * 2; t += 256) {
    const int which = (t >= NN * NH);
    const int idx = which ? t - NN * NH : t;
    const int n = idx >> 2, hh = idx & 3;
    const float* av = which ? adst : asrc;
    const float* wrow = &WhS[n * HID + hh * DD];
    float acc = 0.f;
#pragma unroll
    for (int d = 0; d < DD; ++d) acc += wrow[d] * av[d];
    (which ? dstS : srcS)[idx] = acc;
  }
  __syncthreads();
  const int wave = tid >> 5, lane = tid & 31;
  const float* adjb = adj + (size_t)b * NN * NN;
  const float bnscale = rsqrtf(1.f + 1e-5f);
  for (int task = wave; task < NN * NH; task += 8) {
    const int i = task >> 2, hh = task & 3;
    const int c = hh * DD + lane;
    const float si = srcS[i * NH + hh];
    float m = -3.4e38f, s = 0.f, acc = 0.f;
    for (int j = 0; j < NN; ++j) {
      const float aij = adjb[i * NN + j];           // wave-uniform
      float e = si + dstS[j * NH + hh];
      e = e > 0.f ? e : 0.2f * e;                   // leaky_relu(.,0.2)
      e += 0.5f * aij;
      if (fabsf(aij) < 1e-6f) e = -1e9f;            // mask
      const float mn = fmaxf(m, e);
      const float corr = __expf(m - mn);
      const float p = __expf(e - mn);
      s = s * corr + p;
      acc = acc * corr + p * WhS[j * HID + c];
      m = mn;
    }
    float outv = acc / s;
    float val = outv * (gamma[c] * bnscale) + beta[c];
    val = elu_f(val) + WhS[i * HID + c];
    const size_t oidx = ((size_t)b * NN + i) * HID + c;
    if (residual) val += h[oidx];
    h[oidx] = val;
  }
}

// ---------------------------------------------------------------------------
// Virtual node: hmean = mean_n h[b];  mode 0: vn = vn_emb + hmean
//               mode 1: vn = elu([vn, hmean] @ W_vn + b_vn)
// ---------------------------------------------------------------------------
__global__ __launch_bounds__(128) void k_vn(
    const float* __restrict__ h, float* __restrict__ vn,
    const float* __restrict__ vn_emb, const float* __restrict__ W,
    const float* __restrict__ bias, int mode) {
  __shared__ float cat[2 * HID];
  const int b = blockIdx.x, t = threadIdx.x;
  const float* hb = h + (size_t)b * NN * HID;
  float acc = 0.f;
  for (int n = 0; n < NN; ++n) acc += hb[n * HID + t];
  const float hm = acc * (1.f / (float)NN);
  if (mode == 0) { vn[b * HID + t] = vn_emb[t] + hm; return; }
  cat[t] = vn[b * HID + t];
  cat[HID + t] = hm;
  __syncthreads();
  float o = bias[t];
  for (int k = 0; k < 2 * HID; ++k) o += cat[k] * W[k * HID + t];
  vn[b * HID + t] = elu_f(o);
}

// h += 0.1 * vn (broadcast over N)
__global__ void k_addvn(float* __restrict__ h, const float* __restrict__ vn) {
  const size_t i = (size_t)blockIdx.x * 256 + threadIdx.x;
  const int c = (int)(i & 127);
  const int b = (int)(i / ((size_t)NN * HID));
  h[i] += 0.1f * vn[b * HID + c];
}

// ---------------------------------------------------------------------------
// Pool to 9 networks, node-attention softmax, build flat(1280), LayerNorm,
// write normalized flat to output.  One block per batch item, 128 threads.
// ---------------------------------------------------------------------------
__global__ __launch_bounds__(128) void k_pool(
    const float* __restrict__ h, const float* __restrict__ vn,
    const float* __restrict__ pm, const float* __restrict__ Wna1,
    const float* __restrict__ bna1, const float* __restrict__ Wna2,
    const float* __restrict__ bna2, const float* __restrict__ gln,
    const float* __restrict__ bln, float* __restrict__ flat_out) {
  __shared__ float pooled[9][HID];
  __shared__ float sS[9];
  __shared__ float attnS[9];
  __shared__ float red[128];
  const int b = blockIdx.x, t = threadIdx.x;
  const float* hb = h + (size_t)b * NN * HID;
  for (int k = 0; k < 9; ++k) {
    float acc = 0.f;
    for (int n = 0; n < NN; ++n) acc += pm[n * 9 + k] * hb[n * HID + t];
    pooled[k][t] = acc;
  }
  __syncthreads();
  if (t < 32) {
    for (int k = 0; k < 9; ++k) {
      float y = bna1[t];
      for (int c = 0; c < HID; ++c) y += pooled[k][c] * Wna1[c * 32 + t];
      y = tanhf(y);
      float part = y * Wna2[t];
#pragma unroll
      for (int off = 16; off > 0; off >>= 1) part += __shfl_down(part, off, 32);
      if (t == 0) sS[k] = part + bna2[0];
    }
  }
  __syncthreads();
  if (t == 0) {
    float mx = sS[0];
    for (int k = 1; k < 9; ++k) mx = fmaxf(mx, sS[k]);
    float sum = 0.f, e[9];
    for (int k = 0; k < 9; ++k) { e[k] = __expf(sS[k] - mx); sum += e[k]; }
    for (int k = 0; k < 9; ++k) attnS[k] = e[k] / sum;
  }
  __syncthreads();
  // flat[i]: i = m*128 + t ; m<9 -> pooled*attn, m==9 -> vn
  float vloc[10];
  float lsum = 0.f, lsq = 0.f;
#pragma unroll
  for (int m = 0; m < 10; ++m) {
    const float v = (m < 9) ? pooled[m][t] * attnS[m] : vn[b * HID + t];
    vloc[m] = v; lsum += v; lsq += v * v;
  }
  red[t] = lsum; __syncthreads();
  for (int off = 64; off > 0; off >>= 1) {
    if (t < off) red[t] += red[t + off];
    __syncthreads();
  }
  const float mu = red[0] * (1.f / (float)FLATD);
  __syncthreads();
  red[t] = lsq; __syncthreads();
  for (int off = 64; off > 0; off >>= 1) {
    if (t < off) red[t] += red[t + off];
    __syncthreads();
  }
  const float var = red[0] * (1.f / (float)FLATD) - mu * mu;
  const float rinv = rsqrtf(var + 1e-5f);
#pragma unroll
  for (int m = 0; m < 10; ++m) {
    const int i = m * 128 + t;
    flat_out[(size_t)b * FLATD + i] = (vloc[m] - mu) * rinv * gln[i] + bln[i];
  }
}

// ---------------------------------------------------------------------------
// hid = elu(flat(512,1280) @ W_c1(1280,256) + b_c1) -- WMMA, K chunked by 128
// wave w covers col tiles w and w+8
// ---------------------------------------------------------------------------
__global__ __launch_bounds__(256) void k_c1(
    const float* __restrict__ A, const float* __restrict__ Bm,
    const float* __restrict__ bias, float* __restrict__ C) {
  __shared__ float As[16 * 128];
  const int row0 = blockIdx.x * 16;
  const int tid = threadIdx.x;
  const int wave = tid >> 5, lane = tid & 31;
  const int nl = lane & 15, ml = lane & 15, half = lane >> 4;
  const int n0a = wave * 16, n0b = (wave + 8) * 16;
  v8f acc0 = {}, acc1 = {};
  for (int kc = 0; kc < FLATD; kc += 128) {
    __syncthreads();
    for (int e = tid; e < 16 * 128; e += 256) {
      int m = e >> 7, k = e & 127;
      As[e] = A[(size_t)(row0 + m) * FLATD + kc + k];
    }
    __syncthreads();
    for (int k0 = 0; k0 < 128; k0 += 4) {
      const int ka = k0 + 2 * half;
      v2f a; a.x = As[ml * 128 + ka]; a.y = As[ml * 128 + ka + 1];
      const int kg = kc + ka;
      v2f b0; b0.x = Bm[kg * C1D + n0a + nl]; b0.y = Bm[(kg + 1) * C1D + n0a + nl];
      v2f b1; b1.x = Bm[kg * C1D + n0b + nl]; b1.y = Bm[(kg + 1) * C1D + n0b + nl];
      acc0 = wmma4(a, b0, acc0);
      acc1 = wmma4(a, b1, acc1);
    }
  }
  const int mbase = half * 8;
  const float bea = bias[n0a + nl], beb = bias[n0b + nl];
#pragma unroll
  for (int r = 0; r < 8; ++r) {
    const int row = row0 + mbase + r;
    C[(size_t)row * C1D + n0a + nl] = elu_f(acc0[r] + bea);
    C[(size_t)row * C1D + n0b + nl] = elu_f(acc1[r] + beb);
  }
}

// logits = hid(512,256) @ W_c2(256,2) + b_c2
__global__ void k_c2(const float* __restrict__ hid, const float* __restrict__ W,
                     const float* __restrict__ bias, float* __restrict__ out) {
  const int i = blockIdx.x * 128 + threadIdx.x;
  if (i >= NB * 2) return;
  const int bb = i >> 1, c = i & 1;
  const float* hr = hid + (size_t)bb * C1D;
  float acc = bias[c];
  for (int k = 0; k < C1D; ++k) acc += hr[k] * W[k * 2 + c];
  out[i] = acc;
}

// ---------------------------------------------------------------------------
extern "C" void kernel_launch(void* const* d_in, const int* in_sizes, int n_in,
                              void* d_out, int out_size, void* d_ws, size_t ws_size,
                              hipStream_t stream) {
  const float* x       = (const float*)d_in[0];
  const float* adj     = (const float*)d_in[1];
  const float* W_enc   = (const float*)d_in[2];
  const float* b_enc   = (const float*)d_in[3];
  const float* g_in    = (const float*)d_in[4];
  const float* beta_in = (const float*)d_in[5];
  const float* vn_emb  = (const float*)d_in[6];
  const float* gat_W   = (const float*)d_in[7];
  const float* gat_asrc= (const float*)d_in[8];
  const float* gat_adst= (const float*)d_in[9];
  const float* gat_g   = (const float*)d_in[10];
  const float* gat_b   = (const float*)d_in[11];
  const float* W_vn    = (const float*)d_in[12];
  const float* b_vn    = (const float*)d_in[13];
  const float* W_na1   = (const float*)d_in[14];
  const float* b_na1   = (const float*)d_in[15];
  const float* W_na2   = (const float*)d_in[16];
  const float* b_na2   = (const float*)d_in[17];
  const float* g_ln    = (const float*)d_in[18];
  const float* b_ln    = (const float*)d_in[19];
  const float* W_c1    = (const float*)d_in[20];
  const float* b_c1    = (const float*)d_in[21];
  const float* W_c2    = (const float*)d_in[22];
  const float* b_c2    = (const float*)d_in[23];
  const float* pm      = (const float*)d_in[24];

  float* out = (float*)d_out;
  float* flat_out = out + NB * 2;           // (512,1280) after logits

  float* ws = (float*)d_ws;
  const size_t HSZ = (size_t)BN * HID;      // 7,602,176 floats
  float* h   = ws;
  float* Wh  = ws + HSZ;
  float* vnb = ws + 2 * HSZ;                // 512*128
  float* hid = vnb + NB * HID;              // 512*256

  const int gemm_blocks = BN / 16;          // 3712

  k_encoder<<<gemm_blocks, 256, 0, stream>>>(x, W_enc, b_enc, g_in, beta_in, h);
  k_vn<<<NB, 128, 0, stream>>>(h, vnb, vn_emb, W_vn, b_vn, /*mode=*/0);

  for (int l = 0; l < 3; ++l) {
    k_gemm128<<<gemm_blocks, 256, 0, stream>>>(h, gat_W + (size_t)l * HID * HID, Wh);
    k_attn<<<NB, 256, 0, stream>>>(Wh, adj, gat_asrc + l * DD, gat_adst + l * DD,
                                   gat_g + l * HID, gat_b + l * HID, h,
                                   /*residual=*/(l > 0) ? 1 : 0);
    k_vn<<<NB, 128, 0, stream>>>(h, vnb, vn_emb, W_vn, b_vn, /*mode=*/1);
    if (l < 2)
      k_addvn<<<(int)(HSZ / 256), 256, 0, stream>>>(h, vnb);
  }

  k_pool<<<NB, 128, 0, stream>>>(h, vnb, pm, W_na1, b_na1, W_na2, b_na2,
                                 g_ln, b_ln, flat_out);
  k_c1<<<NB / 16, 256, 0, stream>>>(flat_out, W_c1, b_c1, hid);
  k_c2<<<(NB * 2 + 127) / 128, 128, 0, stream>>>(hid, W_c2, b_c2, out);
}